// InnerProductDecoder_2645699854338
// MI455X (gfx1250) — compile-verified
//
#include <hip/hip_runtime.h>

typedef __attribute__((ext_vector_type(2))) float v2f;
typedef __attribute__((ext_vector_type(8))) float v8f;

#define D_DIM 128
#define K_DIM 64

// ---------------------------------------------------------------------------
// Kernel 1: c_emb[N,128] = c[N,64] @ psi[64,128] using V_WMMA_F32_16X16X4_F32.
// One wave computes a 16x64 tile (4 accumulators of 16x16).
//   A (16x4 f32, ISA 7.12.2): lanes 0-15 -> {K=0,K=1}, lanes 16-31 -> {K=2,K=3}
//   B (4x16 f32, mirrored):   lane n=lane&15, same K split
//   C/D (16x16 f32): VGPR r -> row r (lanes 0-15) / row 8+r (lanes 16-31)
// ---------------------------------------------------------------------------
__global__ __launch_bounds__(256) void cemb_wmma_kernel(
    const float* __restrict__ c,    // [N, 64]
    const float* __restrict__ psi,  // [64, 128]
    float* __restrict__ c_emb,      // [N, 128]
    int Nn) {
  const int wave = (blockIdx.x * blockDim.x + threadIdx.x) >> 5;
  const int lane = threadIdx.x & 31;
  const int mtile = wave >> 1;   // 16-row tile index
  const int dhalf = wave & 1;    // columns 0-63 or 64-127
  if (mtile * 16 >= Nn) return;  // uniform per wave -> EXEC stays all-ones

  const int m  = lane & 15;
  const int kb = (lane >> 4) << 1;          // 0 (lanes 0-15) or 2 (lanes 16-31)
  const int n0 = dhalf * 64 + (lane & 15);  // base output column for this lane

  int arow = mtile * 16 + m;
  if (arow >= Nn) arow = Nn - 1;  // clamp (N=50000 is an exact multiple of 16)
  const float2* __restrict__ crow2 = (const float2*)(c + (size_t)arow * K_DIM);

  v8f acc[4] = {};

#pragma unroll
  for (int k0 = 0; k0 < K_DIM; k0 += 4) {
    // A fragment: two consecutive K values for this lane's row (8B aligned)
    const float2 av = crow2[(k0 + kb) >> 1];
    v2f a;
    a.x = av.x;
    a.y = av.y;

#pragma unroll
    for (int t = 0; t < 4; ++t) {
      v2f b;
      b.x = psi[(size_t)(k0 + kb) * D_DIM + n0 + t * 16];
      b.y = psi[(size_t)(k0 + kb + 1) * D_DIM + n0 + t * 16];
      acc[t] = __builtin_amdgcn_wmma_f32_16x16x4_f32(
          /*neg_a=*/false, a, /*neg_b=*/false, b,
          /*c_mod=*/(short)0, acc[t], /*reuse_a=*/false, /*reuse_b=*/false);
    }
  }

  // Store D: VGPR r holds row (r or 8+r) depending on lane half.
  const int rbase = mtile * 16 + ((lane >> 4) << 3);
  const int col = dhalf * 64 + (lane & 15);
#pragma unroll
  for (int t = 0; t < 4; ++t) {
#pragma unroll
    for (int r = 0; r < 8; ++r) {
      const int row = rbase + r;
      if (row < Nn) c_emb[(size_t)row * D_DIM + col + t * 16] = acc[t][r];
    }
  }
}

// ---------------------------------------------------------------------------
// Kernel 2: per-edge gathered dot products + sigmoids.
// One wave per edge: 32 lanes x float4 = one full 512B row per b128 load
// (4 coalesced row reads per edge, all L2-resident: z+c_emb = 51MB << 192MB L2).
// Butterfly shfl_xor reduction over wave32, lane 0 writes the result.
// ---------------------------------------------------------------------------
__global__ __launch_bounds__(256) void edge_dot_kernel(
    const float* __restrict__ z,       // [N, 128]
    const float* __restrict__ c_emb,   // [N, 128]
    const long long* __restrict__ ei,  // [2, E] int64
    const float* __restrict__ weights, // [1]
    float* __restrict__ out,           // [E]
    int Ee) {
  const int wave = (blockIdx.x * blockDim.x + threadIdx.x) >> 5;
  const int lane = threadIdx.x & 31;
  const int nwaves = (gridDim.x * blockDim.x) >> 5;
  const float w = weights[0];

  for (int e = wave; e < Ee; e += nwaves) {
    const int s = (int)ei[e];
    const int d = (int)ei[(size_t)Ee + e];

    const float4* __restrict__ cs = (const float4*)(c_emb + (size_t)s * D_DIM);
    const float4* __restrict__ zd = (const float4*)(z + (size_t)d * D_DIM);
    const float4* __restrict__ zs = (const float4*)(z + (size_t)s * D_DIM);
    const float4* __restrict__ cd = (const float4*)(c_emb + (size_t)d * D_DIM);

    const float4 a = cs[lane];  // c_emb[src]
    const float4 b = zd[lane];  // z[dst]
    const float4 p = zs[lane];  // z[src]
    const float4 q = cd[lane];  // c_emb[dst]

    float cz = a.x * b.x + a.y * b.y + a.z * b.z + a.w * b.w;
    float zc = p.x * q.x + p.y * q.y + p.z * q.z + p.w * q.w;

#pragma unroll
    for (int off = 16; off > 0; off >>= 1) {
      cz += __shfl_xor(cz, off, 32);
      zc += __shfl_xor(zc, off, 32);
    }

    if (lane == 0) {
      const float s1 = 1.0f / (1.0f + __expf(-w * zc));
      const float s2 = 1.0f / (1.0f + __expf(-w * cz));
      out[e] = 0.5f * (s1 + s2);
    }
  }
}

// ---------------------------------------------------------------------------
extern "C" void kernel_launch(void* const* d_in, const int* in_sizes, int n_in,
                              void* d_out, int out_size, void* d_ws, size_t ws_size,
                              hipStream_t stream) {
  const float* z          = (const float*)d_in[0];      // [N,128]
  const float* c          = (const float*)d_in[1];      // [N,64]
  const float* psi        = (const float*)d_in[2];      // [64,128]
  const float* weights    = (const float*)d_in[3];      // [1]
  const long long* ei     = (const long long*)d_in[4];  // [2,E]
  float* out              = (float*)d_out;              // [E]
  float* c_emb            = (float*)d_ws;               // N*128 floats scratch

  const int N_ = in_sizes[0] / D_DIM;
  const int E_ = out_size;

  // GEMM: 2 waves per 16-row tile (each covers 64 of the 128 cols)
  const int mtiles = (N_ + 15) / 16;
  const int gwaves = mtiles * 2;
  const int gblocks = (gwaves + 7) / 8;  // 8 waves / 256-thread block
  cemb_wmma_kernel<<<gblocks, 256, 0, stream>>>(c, psi, c_emb, N_);

  // Edge kernel: one wave per edge
  const int eblocks = (E_ + 7) / 8;
  edge_dot_kernel<<<eblocks, 256, 0, stream>>>(z, c_emb, ei, weights, out, E_);
}